// CNN_Final_VN_Model_58385785422522
// MI455X (gfx1250) — compile-verified
//
#include <hip/hip_runtime.h>

#define H 64

typedef __attribute__((ext_vector_type(2))) float v2f;
typedef __attribute__((ext_vector_type(8))) float v8f;

__device__ __forceinline__ int fkey(float f) {
    int i = __float_as_int(f);
    return i >= 0 ? i : (i ^ 0x7fffffff);
}
__device__ __forceinline__ float keyf(int k) {
    return __int_as_float(k >= 0 ? k : (k ^ 0x7fffffff));
}

// ---------------------------------------------------------------------------
// Fast path: Y[N x 64] = X[N x 64] @ W[64 x 64]; n % 16 == 0; no guards.
// One wave per 16x16 tile; K=64 fully unrolled (16 WMMAs, straight-line loads).
// ---------------------------------------------------------------------------
__global__ __launch_bounds__(256)
void gemm_wmma_64(const float* __restrict__ X, const float* __restrict__ W,
                  float* __restrict__ Y, int ntiles) {
    int wid = (blockIdx.x * blockDim.x + threadIdx.x) >> 5;
    if (wid >= ntiles) return;               // wave-uniform
    int lane = threadIdx.x & 31;
    int tm = wid >> 2, tn = wid & 3;
    int half = lane >> 4, lr = lane & 15;
    int row = tm * 16 + lr;
    int col = tn * 16 + lr;
    const float* Xr = X + (size_t)row * H + 2 * half;   // A: K=kb,kb+1 contiguous
    const float* Wc = W + (size_t)(2 * half) * H + col; // B: K rows stride H
    v8f c = {};
#pragma unroll
    for (int k0 = 0; k0 < H; k0 += 4) {
        v2f a = *(const v2f*)(Xr + k0);      // global_load_b64
        v2f b;
        b.x = Wc[k0 * H];
        b.y = Wc[k0 * H + H];
        c = __builtin_amdgcn_wmma_f32_16x16x4_f32(false, a, false, b,
                                                  (short)0, c, false, false);
    }
    float* Yp = Y + (size_t)(tm * 16 + half * 8) * H + col;
#pragma unroll
    for (int j = 0; j < 8; ++j) Yp[j * H] = c[j];
}

// ---------------------------------------------------------------------------
// Layer-0 path: Y[N x 64] = X[N x 3] @ W[3 x 64]; K padded 3 -> 4 branch-free.
// ---------------------------------------------------------------------------
__global__ __launch_bounds__(256)
void gemm_wmma_3(const float* __restrict__ X, const float* __restrict__ W,
                 float* __restrict__ Y, int ntiles) {
    int wid = (blockIdx.x * blockDim.x + threadIdx.x) >> 5;
    if (wid >= ntiles) return;
    int lane = threadIdx.x & 31;
    int tm = wid >> 2, tn = wid & 3;
    int half = lane >> 4, lr = lane & 15;
    int row = tm * 16 + lr;
    int col = tn * 16 + lr;
    v2f a, b;
    a.x = X[(size_t)row * 3 + 2 * half];          // K=0 (half 0) / K=2 (half 1)
    float ax1 = X[(size_t)row * 3 + 1];           // always in bounds
    a.y = half ? 0.f : ax1;                       // K=1 / K=3(pad)
    b.x = W[(size_t)(2 * half) * H + col];
    float bx1 = W[H + col];
    b.y = half ? 0.f : bx1;
    v8f c = {};
    c = __builtin_amdgcn_wmma_f32_16x16x4_f32(false, a, false, b,
                                              (short)0, c, false, false);
    float* Yp = Y + (size_t)(tm * 16 + half * 8) * H + col;
#pragma unroll
    for (int j = 0; j < 8; ++j) Yp[j * H] = c[j];
}

// ---------------------------------------------------------------------------
// Generic guarded fallback (only used if shapes don't match the fast paths).
// ---------------------------------------------------------------------------
__global__ __launch_bounds__(256)
void gemm_wmma_gen(const float* __restrict__ X, const float* __restrict__ W,
                   float* __restrict__ Y, int n, int din) {
    int wid  = (blockIdx.x * blockDim.x + threadIdx.x) >> 5;
    int lane = threadIdx.x & 31;
    int ntm  = (n + 15) >> 4;
    if (wid >= ntm * 4) return;
    int tm = wid >> 2, tn = wid & 3;
    int half = lane >> 4, lr = lane & 15;
    int row = tm * 16 + lr;
    int col = tn * 16 + lr;
    v8f c = {};
    for (int k0 = 0; k0 < din; k0 += 4) {
        int kb = k0 + 2 * half;
        v2f a, b;
        a.x = (row < n && kb     < din) ? X[(size_t)row * din + kb]     : 0.f;
        a.y = (row < n && kb + 1 < din) ? X[(size_t)row * din + kb + 1] : 0.f;
        b.x = (kb     < din) ? W[(size_t)kb * H + col]       : 0.f;
        b.y = (kb + 1 < din) ? W[(size_t)(kb + 1) * H + col] : 0.f;
        c = __builtin_amdgcn_wmma_f32_16x16x4_f32(false, a, false, b,
                                                  (short)0, c, false, false);
    }
    int rbase = tm * 16 + half * 8;
#pragma unroll
    for (int j = 0; j < 8; ++j) {
        int r = rbase + j;
        if (r < n) Y[(size_t)r * H + col] = c[j];
    }
}

// ---------------------------------------------------------------------------
// Final projection: Y[N x dout] = Hh[N x 64] @ Wo[64 x dout] + bo (dout=20).
// Branch-free B masking (clamped address * 0/1 mask); n % 16 == 0.
// ---------------------------------------------------------------------------
__global__ __launch_bounds__(256)
void outgemm_wmma(const float* __restrict__ Hh, const float* __restrict__ Wo,
                  const float* __restrict__ bo, float* __restrict__ Y,
                  int ntm, int dout) {
    int wid  = (blockIdx.x * blockDim.x + threadIdx.x) >> 5;
    int ntn  = (dout + 15) >> 4;
    if (wid >= ntm * ntn) return;
    int lane = threadIdx.x & 31;
    int tm = wid / ntn, tn = wid % ntn;
    int half = lane >> 4, lr = lane & 15;
    int row = tm * 16 + lr;
    int col = tn * 16 + lr;
    int   colc  = col < dout ? col : (dout - 1);
    float cmask = col < dout ? 1.f : 0.f;
    const float* Ar = Hh + (size_t)row * H + 2 * half;
    const float* Wc = Wo + (size_t)(2 * half) * dout + colc;
    v8f c = {};
#pragma unroll
    for (int k0 = 0; k0 < H; k0 += 4) {
        v2f a = *(const v2f*)(Ar + k0);
        v2f b;
        b.x = cmask * Wc[k0 * dout];
        b.y = cmask * Wc[k0 * dout + dout];
        c = __builtin_amdgcn_wmma_f32_16x16x4_f32(false, a, false, b,
                                                  (short)0, c, false, false);
    }
    if (col < dout) {
        float bb = bo[col];
        float* Yp = Y + (size_t)(tm * 16 + half * 8) * dout + col;
#pragma unroll
        for (int j = 0; j < 8; ++j) Yp[j * dout] = c[j] + bb;
    }
}

// per-node attention scalars: hs = h.a_s, hd = h.a_d (float4 row loads)
__global__ __launch_bounds__(256)
void node_scalars(const float* __restrict__ Hh, const float* __restrict__ as_,
                  const float* __restrict__ ad_, float* __restrict__ hs,
                  float* __restrict__ hd, int n) {
    int i = blockIdx.x * blockDim.x + threadIdx.x;
    if (i >= n) return;
    const float4* Hp = (const float4*)(Hh + (size_t)i * H);
    float s = 0.f, d = 0.f;
#pragma unroll
    for (int q = 0; q < H / 4; ++q) {
        float4 v = Hp[q];
        s += v.x * as_[4 * q]     + v.y * as_[4 * q + 1]
           + v.z * as_[4 * q + 2] + v.w * as_[4 * q + 3];
        d += v.x * ad_[4 * q]     + v.y * ad_[4 * q + 1]
           + v.z * ad_[4 * q + 2] + v.w * ad_[4 * q + 3];
    }
    hs[i] = s; hd[i] = d;
}

__global__ __launch_bounds__(256)
void layer_init(float* __restrict__ agg, int* __restrict__ mkey,
                float* __restrict__ denom, int n) {
    int i = blockIdx.x * blockDim.x + threadIdx.x;
    if (i < n * H) agg[i] = 0.f;
    if (i < n) { mkey[i] = (int)0x807fffff; denom[i] = 0.f; }  // key(-inf)
}

// pass 1: logits + segment max (int-key atomicMax)
__global__ __launch_bounds__(256)
void edge_logits(const int* __restrict__ esrc, const int* __restrict__ edst,
                 int E_, int n, const float* __restrict__ hs,
                 const float* __restrict__ hd, float* __restrict__ lg,
                 int* __restrict__ mkey) {
    int e = blockIdx.x * blockDim.x + threadIdx.x;
    int ET = E_ + n;
    if (e >= ET) return;
    int s, d;
    if (e < E_) { s = esrc[e]; d = edst[e]; } else { s = d = e - E_; }
    float v = hs[s] + hd[d];
    v = v > 0.f ? v : 0.2f * v;          // LeakyReLU(0.2)
    lg[e] = v;
    atomicMax(&mkey[d], fkey(v));
}

// pass 2: p = exp(lg - m[dst]); denom[dst] += p
__global__ __launch_bounds__(256)
void edge_exp(const int* __restrict__ edst, int E_, int n,
              float* __restrict__ lg, const int* __restrict__ mkey,
              float* __restrict__ denom) {
    int e = blockIdx.x * blockDim.x + threadIdx.x;
    int ET = E_ + n;
    if (e >= ET) return;
    int d = (e < E_) ? edst[e] : (e - E_);
    float p = __expf(lg[e] - keyf(mkey[d]));
    lg[e] = p;
    atomicAdd(&denom[d], p);
}

// pass 3: wave per edge: agg[dst] += (p/denom[dst]) * h[src]
// lane covers features 2*lane, 2*lane+1 -> one b64 load + two f32 atomics
__global__ __launch_bounds__(256)
void edge_scatter(const int* __restrict__ esrc, const int* __restrict__ edst,
                  int E_, int n, const float* __restrict__ p,
                  const float* __restrict__ denom, const float* __restrict__ Hh,
                  float* __restrict__ agg) {
    int wid  = (blockIdx.x * blockDim.x + threadIdx.x) >> 5;
    int lane = threadIdx.x & 31;
    int ET = E_ + n;
    if (wid >= ET) return;
    int s, d;
    if (wid < E_) { s = esrc[wid]; d = edst[wid]; } else { s = d = wid - E_; }
    float alpha = p[wid] / denom[d];
    float2 hv = *(const float2*)(Hh + (size_t)s * H + 2 * lane);
    atomicAdd(&agg[(size_t)d * H + 2 * lane],     alpha * hv.x);
    atomicAdd(&agg[(size_t)d * H + 2 * lane + 1], alpha * hv.y);
}

// out = leaky_relu(agg + bias, 0.01)
__global__ __launch_bounds__(256)
void layer_finalize(float* __restrict__ agg, const float* __restrict__ bias, int n) {
    int i = blockIdx.x * blockDim.x + threadIdx.x;
    if (i >= n * H) return;
    float v = agg[i] + bias[i & (H - 1)];
    agg[i] = v > 0.f ? v : 0.01f * v;
}

__global__ void vn_zero(float* __restrict__ vn) {
    if (threadIdx.x < 8 * H) vn[threadIdx.x] = 0.f;
}

// global_add_pool via LDS accumulation (8 graphs x 64 feats)
__global__ __launch_bounds__(256)
void pool_kernel(const float* __restrict__ Hh, const int* __restrict__ batch,
                 float* __restrict__ vn, int n) {
    __shared__ float acc[8 * H];
    int t = threadIdx.x;
    acc[t] = 0.f; acc[t + 256] = 0.f;
    __syncthreads();
    int k = t & 63;
    int base = blockIdx.x * 1024 + (t >> 6);
    int lim = blockIdx.x * 1024 + 1024; if (lim > n) lim = n;
    for (int i = base; i < lim; i += 4) {
        int g = batch[i];
        atomicAdd(&acc[g * H + k], Hh[(size_t)i * H + k]);   // ds_add_f32
    }
    __syncthreads();
    atomicAdd(&vn[t],       acc[t]);
    atomicAdd(&vn[t + 256], acc[t + 256]);
}

// virtual-node MLP: 4 dense layers on 8x64, one 512-thread block
__global__ __launch_bounds__(512)
void vn_mlp(const float* __restrict__ vn_in, const float* __restrict__ emb,
            const float* __restrict__ w1, const float* __restrict__ b1,
            const float* __restrict__ w2, const float* __restrict__ b2,
            const float* __restrict__ w3, const float* __restrict__ b3,
            const float* __restrict__ w4, const float* __restrict__ b4,
            float* __restrict__ outvn, int dout) {
    __shared__ float buf0[8 * H], buf1[8 * H];
    int t = threadIdx.x;
    int g = t >> 6, k = t & 63;
    buf0[t] = vn_in[t] + emb[k];
    __syncthreads();
    float a = b1[k];
    for (int kk = 0; kk < H; ++kk) a += buf0[g * H + kk] * w1[kk * H + k];
    buf1[t] = a > 0.f ? a : 0.f;
    __syncthreads();
    a = b2[k];
    for (int kk = 0; kk < H; ++kk) a += buf1[g * H + kk] * w2[kk * H + k];
    buf0[t] = a > 0.f ? a : 0.f;
    __syncthreads();
    a = b3[k];
    for (int kk = 0; kk < H; ++kk) a += buf0[g * H + kk] * w3[kk * H + k];
    buf1[t] = a > 0.f ? a : 0.f;
    __syncthreads();
    if (k < dout) {
        a = b4[k];
        for (int kk = 0; kk < H; ++kk) a += buf1[g * H + kk] * w4[kk * dout + k];
        outvn[g * dout + k] = a > 0.f ? a : 0.f;
    }
}

extern "C" void kernel_launch(void* const* d_in, const int* in_sizes, int n_in,
                              void* d_out, int out_size, void* d_ws, size_t ws_size,
                              hipStream_t stream) {
    const float* x      = (const float*)d_in[0];
    const int*   eidx   = (const int*)d_in[1];
    const int*   batch  = (const int*)d_in[2];
    const float* Wl[3]  = {(const float*)d_in[3],  (const float*)d_in[7],  (const float*)d_in[11]};
    const float* asl[3] = {(const float*)d_in[4],  (const float*)d_in[8],  (const float*)d_in[12]};
    const float* adl[3] = {(const float*)d_in[5],  (const float*)d_in[9],  (const float*)d_in[13]};
    const float* bl[3]  = {(const float*)d_in[6],  (const float*)d_in[10], (const float*)d_in[14]};
    const float* vn_emb = (const float*)d_in[15];
    const float* m1w1 = (const float*)d_in[16]; const float* m1b1 = (const float*)d_in[17];
    const float* m1w2 = (const float*)d_in[18]; const float* m1b2 = (const float*)d_in[19];
    const float* mfw1 = (const float*)d_in[20]; const float* mfb1 = (const float*)d_in[21];
    const float* mfw2 = (const float*)d_in[22]; const float* mfb2 = (const float*)d_in[23];
    const float* outw = (const float*)d_in[24]; const float* outb = (const float*)d_in[25];

    const int N_  = in_sizes[2];
    const int E_  = in_sizes[1] / 2;
    const int ET  = E_ + N_;
    const int din = in_sizes[0] / N_;
    const int dout = in_sizes[25];
    const int* esrc = eidx;
    const int* edst = eidx + E_;
    const bool n16 = (N_ % 16) == 0;

    // workspace carve-up
    char* ws = (char*)d_ws;
    float* bufA  = (float*)ws; ws += (size_t)N_ * H * sizeof(float);  // h = X@W
    float* bufB  = (float*)ws; ws += (size_t)N_ * H * sizeof(float);  // agg / layer output
    float* hs    = (float*)ws; ws += (size_t)N_ * sizeof(float);
    float* hd    = (float*)ws; ws += (size_t)N_ * sizeof(float);
    int*   mkey  = (int*)  ws; ws += (size_t)N_ * sizeof(int);
    float* denom = (float*)ws; ws += (size_t)N_ * sizeof(float);
    float* pbuf  = (float*)ws; ws += (size_t)ET * sizeof(float);
    float* vn0   = (float*)ws; ws += 8 * H * sizeof(float);

    const float* Xcur = x;
    for (int l = 0; l < 3; ++l) {
        int dinl  = (l == 0) ? din : H;
        int tiles = ((N_ + 15) / 16) * 4;
        int gemm_grid = (tiles * 32 + 255) / 256;
        if (n16 && dinl == H) {
            gemm_wmma_64<<<gemm_grid, 256, 0, stream>>>(Xcur, Wl[l], bufA, tiles);
        } else if (n16 && dinl == 3) {
            gemm_wmma_3<<<gemm_grid, 256, 0, stream>>>(Xcur, Wl[l], bufA, tiles);
        } else {
            gemm_wmma_gen<<<gemm_grid, 256, 0, stream>>>(Xcur, Wl[l], bufA, N_, dinl);
        }
        node_scalars<<<(N_ + 255) / 256, 256, 0, stream>>>(
            bufA, asl[l], adl[l], hs, hd, N_);
        layer_init<<<((size_t)N_ * H + 255) / 256, 256, 0, stream>>>(
            bufB, mkey, denom, N_);
        edge_logits<<<(ET + 255) / 256, 256, 0, stream>>>(
            esrc, edst, E_, N_, hs, hd, pbuf, mkey);
        edge_exp<<<(ET + 255) / 256, 256, 0, stream>>>(
            edst, E_, N_, pbuf, mkey, denom);
        edge_scatter<<<((size_t)ET * 32 + 255) / 256, 256, 0, stream>>>(
            esrc, edst, E_, N_, pbuf, denom, bufA, bufB);
        layer_finalize<<<((size_t)N_ * H + 255) / 256, 256, 0, stream>>>(
            bufB, bl[l], N_);
        Xcur = bufB;
    }

    vn_zero<<<1, 512, 0, stream>>>(vn0);
    pool_kernel<<<(N_ + 1023) / 1024, 256, 0, stream>>>(bufB, batch, vn0, N_);
    vn_mlp<<<1, 512, 0, stream>>>(vn0, vn_emb, m1w1, m1b1, m1w2, m1b2,
                                  mfw1, mfb1, mfw2, mfb2,
                                  (float*)d_out + (size_t)N_ * dout, dout);

    int ntm = (N_ + 15) / 16;
    int otiles = ntm * ((dout + 15) / 16);
    outgemm_wmma<<<(otiles * 32 + 255) / 256, 256, 0, stream>>>(
        bufB, outw, outb, (float*)d_out, ntm, dout);
}